// GCN_8770323219094
// MI455X (gfx1250) — compile-verified
//
#include <hip/hip_runtime.h>
#include <math.h>

#define NNODES 50000
#define FEAT   256
#define NEDGE  800000
#define NF     (NNODES * FEAT)          // 12,800,000 floats per feature buffer
#define LDSTRIDE 260                    // 256 + 4 pad -> conflict-free A reads

typedef float v2f __attribute__((ext_vector_type(2)));
typedef float v8f __attribute__((ext_vector_type(8)));

// Guaranteed-native fp32 global atomic add (non-returning -> STOREcnt path).
// s_endpgm performs an implicit wait-idle, so no explicit s_wait_storecnt needed.
__device__ __forceinline__ void gatomic_add_f32(float* p, float v) {
    asm volatile("global_atomic_add_f32 %0, %1, off"
                 :
                 : "v"(p), "v"(v)
                 : "memory");
}

// ---------------- small elementwise kernels ----------------

__global__ void k_zero(float* __restrict__ p, int n) {
    int i = blockIdx.x * blockDim.x + threadIdx.x;
    if (i < n) p[i] = 0.0f;
}

__global__ void k_degree(const int* __restrict__ dst, float* __restrict__ deg, int e) {
    int i = blockIdx.x * blockDim.x + threadIdx.x;
    if (i < e) gatomic_add_f32(&deg[dst[i]], 1.0f);
}

// deg -> deg^{-1/2} in place (self-loop adds 1, so deg+1 >= 1 always)
__global__ void k_dinv(float* __restrict__ deg, int n) {
    int i = blockIdx.x * blockDim.x + threadIdx.x;
    if (i < n) deg[i] = rsqrtf(deg[i] + 1.0f);
}

// ---------------- fp32 WMMA GEMM: C[M x 256] = A[M x 256] * W[256 x 256] ----------------
// grid.x = M/16 tiles, block = 512 threads = 16 waves, wave w -> N-tile w.

__global__ __launch_bounds__(512) void k_gemm_wmma(const float* __restrict__ A,
                                                   const float* __restrict__ W,
                                                   float* __restrict__ C) {
    __shared__ float As[16 * LDSTRIDE];

    const int tileM = blockIdx.x;
    const float* Ab = A + (size_t)tileM * 16 * FEAT;

    // stage 16 x 256 fp32 rows of A into LDS (16 KB)
    for (int g = threadIdx.x; g < 16 * FEAT; g += 512) {
        int r = g >> 8, c = g & 255;
        As[r * LDSTRIDE + c] = Ab[g];
    }
    __syncthreads();

    const int lane = threadIdx.x & 31;
    const int wv   = threadIdx.x >> 5;        // 0..15 -> output column tile
    const int row  = lane & 15;               // A fragment row (M)
    const int kk   = (lane >> 4) << 1;        // K sub-offset: lanes 16..31 hold K=2,3
    const int col  = (wv << 4) + (lane & 15); // B fragment column (N)

    v8f acc = {};
    #pragma unroll 4
    for (int k = 0; k < FEAT; k += 4) {
        v2f a, b;
        a.x = As[row * LDSTRIDE + k + kk];
        a.y = As[row * LDSTRIDE + k + kk + 1];
        b.x = W[(size_t)(k + kk)     * FEAT + col];
        b.y = W[(size_t)(k + kk + 1) * FEAT + col];
        acc = __builtin_amdgcn_wmma_f32_16x16x4_f32(
            /*neg_a=*/false, a, /*neg_b=*/false, b,
            /*c_mod=*/(short)0, acc, /*reuse_a=*/false, /*reuse_b=*/false);
    }

    // C/D layout: lanes 0-15 -> M = v, lanes 16-31 -> M = v + 8; N = lane & 15
    const int n     = lane & 15;
    const int mBase = (lane >> 4) << 3;
    float* Cb = C + (size_t)tileM * 16 * FEAT + (wv << 4) + n;
    #pragma unroll
    for (int v = 0; v < 8; v++)
        Cb[(size_t)(mBase + v) * FEAT] = acc[v];
}

// ---------------- aggregation ----------------

// agg[i][:] = h[i][:] * dinv[i]^2   (self-loop term; also initializes agg)
__global__ void k_selfloop(const float* __restrict__ h, const float* __restrict__ dinv,
                           float* __restrict__ agg) {
    int idx  = blockIdx.x * blockDim.x + threadIdx.x;   // one float4 per thread
    int node = idx >> 6;                                // (idx*4) / 256
    float d = dinv[node];
    float s = d * d;
    float4 v = ((const float4*)h)[idx];
    float4 o = make_float4(v.x * s, v.y * s, v.z * s, v.w * s);
    ((float4*)agg)[idx] = o;
}

// one wave per edge: agg[dst] += h[src] * (dinv[src]*dinv[dst])
__global__ __launch_bounds__(256) void k_edge_agg(const int* __restrict__ src,
                                                  const int* __restrict__ dst,
                                                  const float* __restrict__ dinv,
                                                  const float* __restrict__ h,
                                                  float* __restrict__ agg, int e) {
    int wv   = (blockIdx.x * blockDim.x + threadIdx.x) >> 5;
    int lane = threadIdx.x & 31;
    if (wv >= e) return;
    int s = src[wv];
    int d = dst[wv];
    float nrm = dinv[s] * dinv[d];
    const float4* hs = (const float4*)(h + (size_t)s * FEAT);
    float4 v0 = hs[lane * 2];
    float4 v1 = hs[lane * 2 + 1];
    float* ad = agg + (size_t)d * FEAT + lane * 8;
    gatomic_add_f32(ad + 0, v0.x * nrm);
    gatomic_add_f32(ad + 1, v0.y * nrm);
    gatomic_add_f32(ad + 2, v0.z * nrm);
    gatomic_add_f32(ad + 3, v0.w * nrm);
    gatomic_add_f32(ad + 4, v1.x * nrm);
    gatomic_add_f32(ad + 5, v1.y * nrm);
    gatomic_add_f32(ad + 6, v1.z * nrm);
    gatomic_add_f32(ad + 7, v1.w * nrm);
}

// out = relu(agg + bias)
__global__ void k_bias_relu(const float* __restrict__ agg, const float* __restrict__ b,
                            float* __restrict__ out) {
    int idx = blockIdx.x * blockDim.x + threadIdx.x;    // one float4 per thread
    int c   = (idx << 2) & 255;
    float4 v  = ((const float4*)agg)[idx];
    float4 bb = *(const float4*)(b + c);
    float4 o;
    o.x = fmaxf(v.x + bb.x, 0.0f);
    o.y = fmaxf(v.y + bb.y, 0.0f);
    o.z = fmaxf(v.z + bb.z, 0.0f);
    o.w = fmaxf(v.w + bb.w, 0.0f);
    ((float4*)out)[idx] = o;
}

// ---------------- final FC (256 -> 2) + log_softmax, one wave per node ----------------

__global__ __launch_bounds__(256) void k_fc_logsoftmax(const float* __restrict__ h,
                                                       const float* __restrict__ Wfc,
                                                       const float* __restrict__ bfc,
                                                       float* __restrict__ out) {
    int node = (blockIdx.x * blockDim.x + threadIdx.x) >> 5;
    int lane = threadIdx.x & 31;

    const float4* hp = (const float4*)(h + (size_t)node * FEAT);
    float4 a0 = hp[lane * 2];
    float4 a1 = hp[lane * 2 + 1];

    // Wfc is [256][2] row-major; lane covers features f..f+7 -> 16 floats
    const float4* wp = (const float4*)(Wfc + (size_t)(lane * 8) * 2);
    float4 w0 = wp[0], w1 = wp[1], w2 = wp[2], w3 = wp[3];

    float p0 = a0.x * w0.x + a0.y * w0.z + a0.z * w1.x + a0.w * w1.z
             + a1.x * w2.x + a1.y * w2.z + a1.z * w3.x + a1.w * w3.z;
    float p1 = a0.x * w0.y + a0.y * w0.w + a0.z * w1.y + a0.w * w1.w
             + a1.x * w2.y + a1.y * w2.w + a1.z * w3.y + a1.w * w3.w;

    #pragma unroll
    for (int off = 16; off >= 1; off >>= 1) {
        p0 += __shfl_xor(p0, off, 32);
        p1 += __shfl_xor(p1, off, 32);
    }

    if (lane == 0) {
        float l0 = p0 + bfc[0];
        float l1 = p1 + bfc[1];
        float m  = fmaxf(l0, l1);
        float ls = logf(expf(l0 - m) + expf(l1 - m));
        out[(size_t)node * 2 + 0] = l0 - m - ls;
        out[(size_t)node * 2 + 1] = l1 - m - ls;
    }
}

// ---------------- driver ----------------

extern "C" void kernel_launch(void* const* d_in, const int* in_sizes, int n_in,
                              void* d_out, int out_size, void* d_ws, size_t ws_size,
                              hipStream_t stream) {
    (void)n_in; (void)out_size; (void)ws_size;

    const float* x   = (const float*)d_in[0];
    const int*   ei  = (const int*)d_in[1];
    const float* W1  = (const float*)d_in[2];
    const float* b1  = (const float*)d_in[3];
    const float* W2  = (const float*)d_in[4];
    const float* b2  = (const float*)d_in[5];
    const float* Wfc = (const float*)d_in[6];
    const float* bfc = (const float*)d_in[7];
    float*       out = (float*)d_out;

    const int E = in_sizes[1] / 2;      // 800000
    const int* src = ei;
    const int* dst = ei + E;

    float* wsf  = (float*)d_ws;
    float* dinv = wsf;                  // 50000 floats (holds deg first)
    float* h    = wsf + 65536;          // NF floats
    float* agg  = wsf + 65536 + (size_t)NF;       // NF floats
    float* tmp  = wsf + 65536 + (size_t)2 * NF;   // NF floats

    const int nodeBlocks = (NNODES + 255) / 256;        // 196
    const int f4Blocks   = NF / 4 / 256;                // 12500
    const int mTiles     = NNODES / 16;                 // 3125
    const int edgeBlocks = E / 8;                       // 100000 (8 waves/block)

    // degree & normalization
    k_zero<<<nodeBlocks, 256, 0, stream>>>(dinv, NNODES);
    k_degree<<<(E + 255) / 256, 256, 0, stream>>>(dst, dinv, E);
    k_dinv<<<nodeBlocks, 256, 0, stream>>>(dinv, NNODES);

    // layer 1
    k_gemm_wmma<<<mTiles, 512, 0, stream>>>(x, W1, h);
    k_selfloop<<<f4Blocks, 256, 0, stream>>>(h, dinv, agg);
    k_edge_agg<<<edgeBlocks, 256, 0, stream>>>(src, dst, dinv, h, agg, E);
    k_bias_relu<<<f4Blocks, 256, 0, stream>>>(agg, b1, tmp);

    // layer 2
    k_gemm_wmma<<<mTiles, 512, 0, stream>>>(tmp, W2, h);
    k_selfloop<<<f4Blocks, 256, 0, stream>>>(h, dinv, agg);
    k_edge_agg<<<edgeBlocks, 256, 0, stream>>>(src, dst, dinv, h, agg, E);
    k_bias_relu<<<f4Blocks, 256, 0, stream>>>(agg, b2, tmp);

    // classifier + log_softmax
    k_fc_logsoftmax<<<NNODES / 8, 256, 0, stream>>>(tmp, Wfc, bfc, out);
}